// TopKGatingRouter_87978110091809
// MI455X (gfx1250) — compile-verified
//
#include <hip/hip_runtime.h>
#include <hip/hip_bf16.h>

#define HIDDEN      2048
#define NUM_EXPERTS 64
#define TOPK        2

#define WAVES_PER_WG    8
#define TOKENS_PER_WAVE 16
#define TOKENS_PER_WG   (WAVES_PER_WG * TOKENS_PER_WAVE)   // 128
#define LROW            68   // padded LDS row stride (floats): (4*lane+e)%64 conflict-free

typedef float v2f __attribute__((ext_vector_type(2)));
typedef float v8f __attribute__((ext_vector_type(8)));

// ---------------------------------------------------------------------------
// Kernel 0: zero the 128-float reduction workspace (counts[64] | probsum[64])
// ---------------------------------------------------------------------------
__global__ void router_zero_ws(float* ws) {
    ws[threadIdx.x] = 0.0f;
}

// ---------------------------------------------------------------------------
// Kernel 1: gate GEMM (v_wmma_f32_16x16x4_f32) + softmax + top-2 + partials
// ---------------------------------------------------------------------------
__global__ void __launch_bounds__(256)
router_gemm_topk(const float* __restrict__ x, const float* __restrict__ W,
                 float* __restrict__ rw, float* __restrict__ ei,
                 float* __restrict__ g_counts, float* __restrict__ g_probsum,
                 int Ntok)
{
    __shared__ float lds_logits[WAVES_PER_WG][TOKENS_PER_WAVE][LROW];
    __shared__ float wg_probsum[NUM_EXPERTS];
    __shared__ float wg_counts[NUM_EXPERTS];

    const int tid  = threadIdx.x;
    const int wave = tid >> 5;
    const int lane = tid & 31;
    const int l16  = lane & 15;     // A: token row (mod 16) / B: expert col (mod 16)
    const int kh   = lane >> 4;     // K half: lanes 0-15 -> K={0,1}, 16-31 -> K={2,3}

    if (tid < NUM_EXPERTS) { wg_probsum[tid] = 0.0f; wg_counts[tid] = 0.0f; }
    __syncthreads();

    const int token0 = blockIdx.x * TOKENS_PER_WG + wave * TOKENS_PER_WAVE;

    // clamp row so EXEC can stay all-ones through the WMMA loop
    int trow = token0 + l16;
    if (trow > Ntok - 1) trow = Ntok - 1;

    const float* xrow = x + (size_t)trow * HIDDEN + 2 * kh;
    const float* wr0  = W + (size_t)(l16 +  0) * HIDDEN + 2 * kh;
    const float* wr1  = W + (size_t)(l16 + 16) * HIDDEN + 2 * kh;
    const float* wr2  = W + (size_t)(l16 + 32) * HIDDEN + 2 * kh;
    const float* wr3  = W + (size_t)(l16 + 48) * HIDDEN + 2 * kh;

    v8f acc0 = {}, acc1 = {}, acc2 = {}, acc3 = {};

    // D = A(16x4) * B(4x16) + C, K advanced 4 per step; 4 expert tiles
    for (int k = 0; k < HIDDEN; k += 4) {
        v2f a  = *(const v2f*)(xrow + k);
        v2f b0 = *(const v2f*)(wr0 + k);
        v2f b1 = *(const v2f*)(wr1 + k);
        v2f b2 = *(const v2f*)(wr2 + k);
        v2f b3 = *(const v2f*)(wr3 + k);
        acc0 = __builtin_amdgcn_wmma_f32_16x16x4_f32(false, a, false, b0, (short)0, acc0, false, false);
        acc1 = __builtin_amdgcn_wmma_f32_16x16x4_f32(false, a, false, b1, (short)0, acc1, false, false);
        acc2 = __builtin_amdgcn_wmma_f32_16x16x4_f32(false, a, false, b2, (short)0, acc2, false, false);
        acc3 = __builtin_amdgcn_wmma_f32_16x16x4_f32(false, a, false, b3, (short)0, acc3, false, false);
    }

    // Spill accumulators: VGPR r holds row (r + 8*kh), column l16 + 16*tile
    #pragma unroll
    for (int r = 0; r < 8; ++r) {
        const int m = r + 8 * kh;
        lds_logits[wave][m][l16 +  0] = acc0[r];
        lds_logits[wave][m][l16 + 16] = acc1[r];
        lds_logits[wave][m][l16 + 32] = acc2[r];
        lds_logits[wave][m][l16 + 48] = acc3[r];
    }
    __syncthreads();

    // Lanes 0-15 of each wave: per-token softmax over 64 experts + top-2
    if (lane < 16) {
        const int tok = token0 + lane;
        if (tok < Ntok) {
            const float* row = &lds_logits[wave][lane][0];

            float mx = -INFINITY;
            #pragma unroll 8
            for (int e = 0; e < NUM_EXPERTS; ++e) mx = fmaxf(mx, row[e]);

            float s = 0.0f;
            #pragma unroll 8
            for (int e = 0; e < NUM_EXPERTS; ++e) s += __expf(row[e] - mx);
            const float inv = 1.0f / s;

            float p1 = -1.0f, p2 = -1.0f;
            int   i1 = 0,     i2 = 0;
            for (int e = 0; e < NUM_EXPERTS; ++e) {
                const float p = __expf(row[e] - mx) * inv;
                atomicAdd(&wg_probsum[e], p);
                if (p > p1)      { p2 = p1; i2 = i1; p1 = p; i1 = e; }
                else if (p > p2) { p2 = p;  i2 = e; }
            }

            const float denom = p1 + p2 + 1e-6f;
            rw[tok * 2 + 0] = p1 / denom;
            rw[tok * 2 + 1] = p2 / denom;
            ei[tok * 2 + 0] = (float)i1;
            ei[tok * 2 + 1] = (float)i2;

            atomicAdd(&wg_counts[i1], 1.0f);
            atomicAdd(&wg_counts[i2], 1.0f);
        }
    }
    __syncthreads();

    // One global atomic per expert per workgroup
    if (tid < NUM_EXPERTS) {
        atomicAdd(&g_probsum[tid], wg_probsum[tid]);
        atomicAdd(&g_counts[tid],  wg_counts[tid]);
    }
}

// ---------------------------------------------------------------------------
// Kernel 2: aux loss + expert_counts finalize (1 block, 64 threads)
// ---------------------------------------------------------------------------
__global__ void router_finalize(const float* __restrict__ g_counts,
                                const float* __restrict__ g_probsum,
                                float* __restrict__ aux_out,
                                float* __restrict__ counts_out,
                                float invN)
{
    __shared__ float red[NUM_EXPERTS];
    const int e = threadIdx.x;

    const float c  = g_counts[e];
    const float ps = g_probsum[e];
    counts_out[e] = c;

    const float fa = c  * invN;   // assign_per_expert
    const float fp = ps * invN;   // prob_per_expert
    red[e] = fp * fp + fa * fa;
    __syncthreads();

    for (int s = NUM_EXPERTS / 2; s > 0; s >>= 1) {
        if (e < s) red[e] += red[e + s];
        __syncthreads();
    }
    if (e == 0) aux_out[0] = red[0] * (float)NUM_EXPERTS;
}

// ---------------------------------------------------------------------------
extern "C" void kernel_launch(void* const* d_in, const int* in_sizes, int n_in,
                              void* d_out, int out_size, void* d_ws, size_t ws_size,
                              hipStream_t stream)
{
    const float* x = (const float*)d_in[0];   // [B,S,H] f32
    const float* W = (const float*)d_in[1];   // [E,H]   f32

    const int Ntok = in_sizes[0] / HIDDEN;    // 16384

    float* out      = (float*)d_out;
    float* rw       = out;                          // [N*2] routing weights
    float* ei       = out + (size_t)Ntok * TOPK;    // [N*2] indices (as float)
    float* aux      = out + (size_t)Ntok * TOPK * 2;      // [1]
    float* cnt_out  = aux + 1;                            // [64]

    float* ws       = (float*)d_ws;
    float* g_counts  = ws;                 // [64]
    float* g_probsum = ws + NUM_EXPERTS;   // [64]

    router_zero_ws<<<1, 2 * NUM_EXPERTS, 0, stream>>>(ws);

    const int grid = (Ntok + TOKENS_PER_WG - 1) / TOKENS_PER_WG;  // 128
    router_gemm_topk<<<grid, 256, 0, stream>>>(x, W, rw, ei, g_counts, g_probsum, Ntok);

    router_finalize<<<1, NUM_EXPERTS, 0, stream>>>(g_counts, g_probsum, aux, cnt_out,
                                                   1.0f / (float)Ntok);
}